// CharNGramVectorizer_45011257262280
// MI455X (gfx1250) — compile-verified
//
#include <hip/hip_runtime.h>
#include <hip/hip_bf16.h>

#define SLEN  512
#define VOCAB 3875          // 5^3 + 5^4 + 5^5
#define NG    1527          // 510 + 509 + 508 windows per row
#define BLK   256

#if __has_builtin(__builtin_amdgcn_global_load_async_to_lds_b64) && \
    __has_builtin(__builtin_amdgcn_global_store_async_from_lds_b32)
#define USE_ASYNC 1
#else
#define USE_ASYNC 0
#endif

typedef int v2i __attribute__((ext_vector_type(2)));
#define GAS __attribute__((address_space(1)))   // global
#define LAS __attribute__((address_space(3)))   // LDS

__device__ __forceinline__ void wait_asynccnt0() {
#if __has_builtin(__builtin_amdgcn_s_wait_asynccnt)
    __builtin_amdgcn_s_wait_asynccnt(0);
#else
    asm volatile("s_wait_asynccnt 0" ::: "memory");
#endif
}

__global__ __launch_bounds__(BLK)
void ngram_presence_kernel(const int* __restrict__ tokens,
                           const float* __restrict__ values,
                           float* __restrict__ out)
{
    __shared__ int tok[SLEN];              // 2 KB
    __shared__ int pres[VOCAB + 1];        // 15.5 KB, float bit patterns

    const int tid = threadIdx.x;
    const int b   = blockIdx.x;

    const int*   tokens_row = tokens + (size_t)b * SLEN;
    const float* vals_row   = values + (size_t)b * NG;
    float*       out_row    = out    + (size_t)b * VOCAB;

    // ---- stage tokens row (2 KB) into LDS via async data mover ----
#if USE_ASYNC
    // 256 threads x 8B = 2048B, one b64 async transfer per lane
    __builtin_amdgcn_global_load_async_to_lds_b64(
        (GAS v2i*)(tokens_row + tid * 2),
        (LAS v2i*)(&tok[tid * 2]),
        0, 0);
#else
    tok[2 * tid]     = tokens_row[2 * tid];
    tok[2 * tid + 1] = tokens_row[2 * tid + 1];
#endif

    // ---- zero presence table while the async load is in flight ----
    for (int i = tid; i < VOCAB; i += BLK) pres[i] = 0;

#if USE_ASYNC
    wait_asynccnt0();
#endif
    __syncthreads();

    // ---- scatter-max n-gram hits into LDS ----
    // vocab layout: 3-grams [0,125), 4-grams [125,750), 5-grams [750,3875)
    for (int p = tid; p < SLEN - 2; p += BLK) {          // 510 3-gram windows
        const int t0 = tok[p], t1 = tok[p + 1], t2 = tok[p + 2];
        const int h3 = t0 * 25 + t1 * 5 + t2;
        atomicMax(&pres[h3], __float_as_int(vals_row[p]));
        if (p < SLEN - 3) {                              // 509 4-gram windows
            const int h4 = h3 * 5 + tok[p + 3];
            atomicMax(&pres[125 + h4], __float_as_int(vals_row[510 + p]));
            if (p < SLEN - 4) {                          // 508 5-gram windows
                const int h5 = h4 * 5 + tok[p + 4];
                atomicMax(&pres[750 + h5], __float_as_int(vals_row[1019 + p]));
            }
        }
    }

    __syncthreads();

    // ---- stream the finished row LDS -> global via async data mover ----
#if USE_ASYNC
    #pragma unroll
    for (int k = 0; k < (VOCAB + BLK - 1) / BLK; ++k) {  // 16 rounds
        const int c = tid + k * BLK;
        if (c < VOCAB) {
            __builtin_amdgcn_global_store_async_from_lds_b32(
                (GAS int*)(out_row + c),
                (LAS int*)(&pres[c]),
                0, 0);
        }
    }
    wait_asynccnt0();
#else
    for (int c = tid; c < VOCAB; c += BLK)
        out_row[c] = __int_as_float(pres[c]);
#endif
}

extern "C" void kernel_launch(void* const* d_in, const int* in_sizes, int n_in,
                              void* d_out, int out_size, void* d_ws, size_t ws_size,
                              hipStream_t stream) {
    (void)n_in; (void)d_ws; (void)ws_size; (void)out_size;
    const int*   tokens = (const int*)d_in[0];
    const float* values = (const float*)d_in[1];
    float*       out    = (float*)d_out;

    const int B = in_sizes[0] / SLEN;     // 4096
    ngram_presence_kernel<<<B, BLK, 0, stream>>>(tokens, values, out);
}